// Decimate_48490180771959
// MI455X (gfx1250) — compile-verified
//
#include <hip/hip_runtime.h>

typedef __attribute__((ext_vector_type(2))) float v2f;
typedef __attribute__((ext_vector_type(8))) float v8f;

#define T_LEN   1048576
#define N_TAPS  129
#define QDEC    4
#define ROWS    16
#define OUT_COLS (T_LEN / QDEC)          // 262144
#define K_CHUNKS 48                       // ceil((4*15 + 129)/4) -> 192 / 4
#define WAVES_PER_BLOCK 8
#define TILES_PER_WAVE  4
#define TILES_TOTAL (OUT_COLS / 16)       // 16384
#define GRID_BLOCKS (TILES_TOTAL / (WAVES_PER_BLOCK * TILES_PER_WAVE)) // 512

// D = A x B + C via V_WMMA_F32_16X16X4_F32 (full fp32 matrix path, wave32).
//   A[r][n] = x[r][4*m0 - 64 + n]          (16 x 192 window, chunks of K=4)
//   B[n][j] = k[n - 4j]                    (Toeplitz; constant -> LDS, fragment layout)
//   D[r][j] = out[r][m0 + j]
// Two interleaved accumulator chains so back-to-back WMMAs pipeline instead of
// serializing on the D->C dependency.
__global__ __launch_bounds__(256) void decimate_wmma_f32(
    const float* __restrict__ x,
    const float* __restrict__ kern,
    float* __restrict__ out)
{
    // B fragments pre-swizzled per lane: entry [c*32 + lane] = (VGPR0, VGPR1).
    //   VGPR0: lanes 0-15 -> K=4c+0, lanes 16-31 -> K=4c+2 ; VGPR1: +1.
    __shared__ v2f ldsB[K_CHUNKS * 32];

    for (int e = threadIdx.x; e < K_CHUNKS * 32; e += blockDim.x) {
        int c  = e >> 5;
        int L  = e & 31;
        int lo = L & 15;
        int hi = L >> 4;
        int k0 = 4 * c + 2 * hi - 4 * lo;      // B[K][N] = k[K - 4N]
        int k1 = k0 + 1;
        float b0 = (k0 >= 0 && k0 < N_TAPS) ? kern[k0] : 0.0f;
        float b1 = (k1 >= 0 && k1 < N_TAPS) ? kern[k1] : 0.0f;
        v2f b; b.x = b0; b.y = b1;
        ldsB[e] = b;
    }
    __syncthreads();

    const int lane = threadIdx.x & 31;
    const int wave = threadIdx.x >> 5;
    const int lo   = lane & 15;
    const int hi   = lane >> 4;

    // A fragment: lane handles matrix row M = lo; hi selects K-subpair (+2).
    const float* xrow = x + (size_t)lo * T_LEN;

    const int waveGlobal = blockIdx.x * WAVES_PER_BLOCK + wave;

    for (int t = 0; t < TILES_PER_WAVE; ++t) {
        const int tile = waveGlobal * TILES_PER_WAVE + t;
        const int m0   = tile << 4;                 // first output column of tile
        const int base = m0 * QDEC - (N_TAPS / 2);  // window start in x (may be <0)

        v8f acc0 = {};
        v8f acc1 = {};

        const bool interior = (base >= 0) && (base + 4 * K_CHUNKS - 1 < T_LEN);
        if (interior) {
            const float* p = xrow + base + 2 * hi;
#pragma unroll
            for (int c = 0; c < K_CHUNKS; c += 2) {
                v2f a0 = *(const v2f*)(p + 4 * c);           // global_load_b64
                v2f a1 = *(const v2f*)(p + 4 * (c + 1));
                v2f b0 = ldsB[c * 32 + lane];                // ds loads (pairable)
                v2f b1 = ldsB[(c + 1) * 32 + lane];
                acc0 = __builtin_amdgcn_wmma_f32_16x16x4_f32(
                        false, a0, false, b0, (short)0, acc0, false, false);
                acc1 = __builtin_amdgcn_wmma_f32_16x16x4_f32(
                        false, a1, false, b1, (short)0, acc1, false, false);
            }
        } else {
            // Edge tiles (first/last): reflect left (i<0 -> -i), zero right (i>=T).
            // Branchless: clamp the address, load unconditionally, mask the value.
#pragma unroll 4
            for (int c = 0; c < K_CHUNKS; ++c) {
                int i0 = base + 4 * c + 2 * hi;
                int i1 = i0 + 1;
                int r0 = (i0 < 0) ? -i0 : i0;                // reflect
                int r1 = (i1 < 0) ? -i1 : i1;
                bool v0 = (r0 < T_LEN);
                bool v1 = (r1 < T_LEN);
                float a0 = xrow[v0 ? r0 : 0];                // safe unconditional load
                float a1 = xrow[v1 ? r1 : 0];
                a0 = v0 ? a0 : 0.0f;                         // v_cndmask, no branch
                a1 = v1 ? a1 : 0.0f;
                v2f a; a.x = a0; a.y = a1;
                v2f b = ldsB[c * 32 + lane];
                if (c & 1) {
                    acc1 = __builtin_amdgcn_wmma_f32_16x16x4_f32(
                            false, a, false, b, (short)0, acc1, false, false);
                } else {
                    acc0 = __builtin_amdgcn_wmma_f32_16x16x4_f32(
                            false, a, false, b, (short)0, acc0, false, false);
                }
            }
        }

        v8f acc = acc0 + acc1;

        // D layout: acc[v] on lane L -> out[v + 8*hi][m0 + lo]
        float* obase = out + (size_t)(8 * hi) * OUT_COLS + (size_t)m0 + lo;
#pragma unroll
        for (int v = 0; v < 8; ++v) {
            obase[(size_t)v * OUT_COLS] = acc[v];
        }
    }
}

extern "C" void kernel_launch(void* const* d_in, const int* in_sizes, int n_in,
                              void* d_out, int out_size, void* d_ws, size_t ws_size,
                              hipStream_t stream)
{
    (void)in_sizes; (void)n_in; (void)d_ws; (void)ws_size; (void)out_size;
    const float* x    = (const float*)d_in[0];
    const float* kern = (const float*)d_in[1];
    // d_in[2] is q (==4), baked into the kernel as QDEC.
    float* out = (float*)d_out;

    hipLaunchKernelGGL(decimate_wmma_f32,
                       dim3(GRID_BLOCKS), dim3(WAVES_PER_BLOCK * 32), 0, stream,
                       x, kern, out);
}